// Net_38826504355941
// MI455X (gfx1250) — compile-verified
//
#include <hip/hip_runtime.h>

typedef __attribute__((ext_vector_type(2))) float v2f;
typedef __attribute__((ext_vector_type(8))) float v8f;

static constexpr int    NN   = 100000;      // nodes
static constexpr int    NE   = 1000000;     // edges
static constexpr int    DIN  = 128;
static constexpr int    HDIM = 512;
static constexpr int    DOUT = 128;

// ---------------- workspace layout (floats) ----------------
static constexpr size_t MSG_ELEMS = (size_t)NN * DIN;        // 12,800,000
static constexpr size_t DEG_OFF   = MSG_ELEMS;
static constexpr size_t DEG_PAD   = 102400;                  // deg + pad, keeps alignment
static constexpr size_t H1_OFF    = DEG_OFF + DEG_PAD;
static constexpr size_t H_ELEMS   = (size_t)NN * HDIM;       // 51,200,000
static constexpr size_t H2_OFF    = H1_OFF + H_ELEMS;

// ---------------- zero init ----------------
__global__ void zero_f4(float4* __restrict__ p, size_t n4) {
    size_t i = (size_t)blockIdx.x * blockDim.x + threadIdx.x;
    size_t stride = (size_t)gridDim.x * blockDim.x;
    float4 z = {0.f, 0.f, 0.f, 0.f};
    for (; i < n4; i += stride) p[i] = z;
}

// ---------------- edge scatter: msg[dst] += feat[src]; deg[dst] += 1 ----------------
// One wave per edge (32 lanes x 4 floats = 128 features); atomics resolve in L2
// (msg_sum is 51 MB, well inside the 192 MB L2).
__global__ __launch_bounds__(256)
void scatter_edges(const float* __restrict__ feats, const int* __restrict__ src,
                   const int* __restrict__ dst, float* __restrict__ msg,
                   float* __restrict__ deg, int nEdges) {
    int idx  = blockIdx.x * blockDim.x + threadIdx.x;
    int e    = idx >> 5;
    int lane = idx & 31;
    if (e >= nEdges) return;
    int s = src[e];
    int d = dst[e];
    const float4 v = *(const float4*)(feats + (size_t)s * DIN + lane * 4);
    float* out = msg + (size_t)d * DIN + lane * 4;
    atomicAdd(out + 0, v.x);
    atomicAdd(out + 1, v.y);
    atomicAdd(out + 2, v.z);
    atomicAdd(out + 3, v.w);
    if (lane == 0) atomicAdd(deg + d, 1.0f);
}

// ---------------- mean normalize in place ----------------
__global__ __launch_bounds__(256)
void normalize_mean(float* __restrict__ msg, const float* __restrict__ deg, int nNodes) {
    int idx  = blockIdx.x * blockDim.x + threadIdx.x;
    int node = idx >> 5;
    int lane = idx & 31;
    if (node >= nNodes) return;
    float inv = 1.0f / fmaxf(deg[node], 1.0f);
    float4* p = (float4*)(msg + (size_t)node * DIN + lane * 4);
    float4 v = *p;
    v.x *= inv; v.y *= inv; v.z *= inv; v.w *= inv;
    *p = v;
}

// ---------------- fp32 WMMA GEMM: C[M,N] = relu?(A[M,K] @ B[K,N] + bias) ----------------
// Block = 256 threads (8 wave32s) owns 32 rows x all N columns.
// A is staged per 32-wide K-block in LDS (padded stride -> conflict-free).
// Each wave computes a 32M x (16*TPI)N register tile: the B fragment is reused
// by both 16-row strips and the A fragment by both column tiles, so the inner
// loop costs 1 global_load_b32 + 0.5 ds_load_b64 per v_wmma_f32_16x16x4_f32.
template <int K, int N, bool RELU>
__global__ __launch_bounds__(256)
void gemm_wmma_f32(const float* __restrict__ A, const float* __restrict__ B,
                   const float* __restrict__ bias, float* __restrict__ C) {
    constexpr int KB     = 32;           // K-block (8 WMMA steps)
    constexpr int SSTR   = KB + 4;       // LDS row stride: 16B-aligned, bank-spread
    constexpr int NTILES = N / 16;
    constexpr int TPI    = (NTILES >= 16) ? 2 : 1;  // column tiles per wave per pass
    constexpr int OUTER  = NTILES / (8 * TPI);      // N=512 -> 2 passes; N=128 -> 1

    __shared__ float sA[32 * SSTR];

    const int row0 = blockIdx.x * 32;
    const int tid  = threadIdx.x;
    const int lane = tid & 31;
    const int wave = tid >> 5;
    const int half = lane >> 4;          // 0: K pair {0,1}; 1: K pair {2,3}
    const int l16  = lane & 15;

    const float* gA = A + (size_t)row0 * K;
    // staging assignment: thread -> (row, 4-float chunk)
    const int st_r  = tid >> 3;          // 0..31
    const int st_c  = (tid & 7) * 4;     // 0..28

    for (int oi = 0; oi < OUTER; ++oi) {
        int          col[TPI];
        const float* bp[TPI];
        v8f          acc[TPI][2];
#pragma unroll
        for (int j = 0; j < TPI; ++j) {
            col[j] = ((oi * TPI + j) * 8 + wave) * 16;
            bp[j]  = B + (size_t)(half * 2) * N + col[j] + l16;
            acc[j][0] = (v8f){};
            acc[j][1] = (v8f){};
        }

        for (int kb = 0; kb < K; kb += KB) {
            __syncthreads();
            *(float4*)&sA[st_r * SSTR + st_c] =
                *(const float4*)(gA + (size_t)st_r * K + kb + st_c);
            __syncthreads();

            const float* ap = &sA[l16 * SSTR + half * 2];
#pragma unroll
            for (int kk = 0; kk < KB; kk += 4) {
                v2f a0 = *(const v2f*)(ap);                 // strip 0 (rows 0..15)
                v2f a1 = *(const v2f*)(ap + 16 * SSTR);     // strip 1 (rows 16..31)
#pragma unroll
                for (int j = 0; j < TPI; ++j) {
                    v2f b;
                    b.x = bp[j][0];                          // B[k  ][col]
                    b.y = bp[j][N];                          // B[k+1][col]
                    acc[j][0] = __builtin_amdgcn_wmma_f32_16x16x4_f32(
                        false, a0, false, b, (short)0, acc[j][0], false, false);
                    acc[j][1] = __builtin_amdgcn_wmma_f32_16x16x4_f32(
                        false, a1, false, b, (short)0, acc[j][1], false, false);
                    bp[j] += (size_t)4 * N;
                }
                ap += 4;
            }
        }

        // epilogue: bias (+ReLU) and store both strips of each tile
#pragma unroll
        for (int j = 0; j < TPI; ++j) {
            const float bv = bias[col[j] + l16];
#pragma unroll
            for (int s = 0; s < 2; ++s) {
                float* cp = C + (size_t)(row0 + s * 16 + half * 8) * N + col[j] + l16;
#pragma unroll
                for (int r = 0; r < 8; ++r) {
                    float v = acc[j][s][r] + bv;
                    if (RELU) v = fmaxf(v, 0.0f);
                    cp[(size_t)r * N] = v;
                }
            }
        }
    }
}

extern "C" void kernel_launch(void* const* d_in, const int* in_sizes, int n_in,
                              void* d_out, int out_size, void* d_ws, size_t ws_size,
                              hipStream_t stream) {
    const float* feats = (const float*)d_in[0];
    const int*   src   = (const int*)d_in[1];
    const int*   dst   = (const int*)d_in[2];
    const float* W1    = (const float*)d_in[3];
    const float* b1    = (const float*)d_in[4];
    const float* Wh    = (const float*)d_in[5];   // [2, 512, 512]
    const float* bh    = (const float*)d_in[6];   // [2, 512]
    const float* Wout  = (const float*)d_in[7];
    const float* bout  = (const float*)d_in[8];
    float*       out   = (float*)d_out;

    float* ws  = (float*)d_ws;
    float* msg = ws;                 // [NN, 128], becomes h0 after normalize
    float* deg = ws + DEG_OFF;       // [NN]
    float* h1  = ws + H1_OFF;        // [NN, 512]
    float* h2  = ws + H2_OFF;        // [NN, 512]

    // 1) zero msg + deg
    {
        size_t n4 = (MSG_ELEMS + (size_t)NN) / 4;   // 12,900,000 / 4
        int blocks = (int)((n4 + 255) / 256);
        if (blocks > 65535) blocks = 65535;
        zero_f4<<<blocks, 256, 0, stream>>>((float4*)msg, n4);
    }

    // 2) scatter-add features over edges + degree
    {
        long long threads = (long long)NE * 32;
        int blocks = (int)((threads + 255) / 256);
        scatter_edges<<<blocks, 256, 0, stream>>>(feats, src, dst, msg, deg, NE);
    }

    // 3) mean normalize -> h0 (in place in msg)
    {
        long long threads = (long long)NN * 32;
        int blocks = (int)((threads + 255) / 256);
        normalize_mean<<<blocks, 256, 0, stream>>>(msg, deg, NN);
    }

    // 4) MLP stack: 32-row panels per block, WMMA fp32 matrix ops
    const int gblocks = NN / 32;     // 3125
    gemm_wmma_f32<DIN,  HDIM, true ><<<gblocks, 256, 0, stream>>>(msg, W1, b1, h1);
    gemm_wmma_f32<HDIM, HDIM, true ><<<gblocks, 256, 0, stream>>>(h1, Wh,               bh,        h2);
    gemm_wmma_f32<HDIM, HDIM, true ><<<gblocks, 256, 0, stream>>>(h2, Wh + HDIM * HDIM, bh + HDIM, h1);
    gemm_wmma_f32<HDIM, DOUT, false><<<gblocks, 256, 0, stream>>>(h1, Wout, bout, out);
}